// GedLayer_23338852287043
// MI455X (gfx1250) — compile-verified
//
#include <hip/hip_runtime.h>
#include <hip/hip_bf16.h>

typedef __attribute__((ext_vector_type(2))) float v2f;
typedef __attribute__((ext_vector_type(8))) float v8f;

#define NLAB 10
#define ELAB 4
#define NG   79
#define NP   80          // padded order (n+1)
#define NN   6400        // NP*NP
#define NODE_INS_DEL 0.03f
#define SINK_IT 10
#define NTHREADS 512
#define NWAVES  16

__launch_bounds__(NTHREADS)
__global__ void ged_kernel(const int* __restrict__ A1,
                           const int* __restrict__ A2,
                           const int* __restrict__ lab1,
                           const int* __restrict__ lab2,
                           const float* __restrict__ node_w,
                           const float* __restrict__ edge_w,
                           float* __restrict__ out)
{
    __shared__ float Vs[NN];                 // Sinkhorn matrix V (80x80)
    __shared__ float Xs[NN];                 // X_a = V * U_a
    __shared__ unsigned char A2p[NN];        // padded graph-2 labels
    __shared__ float tableL[25];             // 5x5 edge-pair cost table
    __shared__ float ncost[NLAB * NLAB];     // 10x10 node cost table
    __shared__ float rsum[NP];
    __shared__ float csum[NP];
    __shared__ float wq[NWAVES], wdg[NWAVES], wcv[NWAVES];

    const int tid  = threadIdx.x;
    const int lane = tid & 31;
    const int wave = tid >> 5;
    const int m    = lane & 15;   // M-row for A-frag / N-col for B,C,D frags
    const int h    = lane >> 4;   // lane half

    // ---- 5x5 edge table: table[a,b] ----
    if (tid < 25) {
        int a = tid / 5, b = tid % 5;
        float v;
        if (a == 0 && b == 0)      v = 0.0f;
        else if (a == 0 || b == 0) v = fmaxf(edge_w[ELAB * (ELAB - 1) / 2], 0.0f); // ins/del
        else if (a == b)           v = 0.0f;
        else {
            int p = (a < b ? a : b) - 1, q = (a < b ? b : a) - 1;
            int idx = p * ELAB - p * (p + 1) / 2 + (q - p - 1);
            v = fmaxf(edge_w[idx], 0.0f);
        }
        tableL[tid] = v;
    }
    // ---- 10x10 node cost table ----
    if (tid < NLAB * NLAB) {
        int p = tid / NLAB, q = tid % NLAB;
        float v = 0.0f;
        if (p != q) {
            int pm = p < q ? p : q, qm = p < q ? q : p;
            int idx = pm * NLAB - pm * (pm + 1) / 2 + (qm - pm - 1);
            v = fmaxf(node_w[idx], 0.0f);
        }
        ncost[tid] = v;
    }
    // ---- padded graph-2 adjacency labels ----
    for (int idx = tid; idx < NN; idx += NTHREADS) {
        int k = idx / NP, l = idx % NP;
        A2p[idx] = (k < NG && l < NG) ? (unsigned char)A2[k * NG + l] : 0;
    }
    __syncthreads();

    // ---- V = exp(-0.5 * c) ----
    for (int idx = tid; idx < NN; idx += NTHREADS) {
        int i = idx / NP, j = idx % NP;
        float cij;
        if (i < NG && j < NG)           cij = ncost[lab1[i] * NLAB + lab2[j]];
        else if (i == NG && j == NG)    cij = 0.0f;
        else                            cij = NODE_INS_DEL;
        Vs[idx] = expf(-0.5f * cij);
    }
    __syncthreads();

    // ---- Sinkhorn: 10 x (row-normalize, col-normalize) ----
    for (int it = 0; it < SINK_IT; ++it) {
        if (tid < NP) {
            float s = 0.0f;
            for (int j = 0; j < NP; ++j) s += Vs[tid * NP + j];
            rsum[tid] = s;
        }
        __syncthreads();
        for (int idx = tid; idx < NN; idx += NTHREADS) Vs[idx] /= rsum[idx / NP];
        __syncthreads();
        if (tid < NP) {
            float s = 0.0f;
            for (int r = 0; r < NP; ++r) s += Vs[r * NP + tid];
            csum[tid] = s;
        }
        __syncthreads();
        for (int idx = tid; idx < NN; idx += NTHREADS) Vs[idx] /= csum[idx % NP];
        __syncthreads();
    }

    // ---- quadratic form: vT C v = sum_a < indicator(A1==a) , (V U_a V^T)^T > ----
    float quad = 0.0f;
    for (int a = 0; a < 5; ++a) {
        const float* trow = &tableL[a * 5];

        // Phase A: X_a = V * U_a,  U_a[k,l] = trow[A2p[k,l]]  (gathered at load)
        for (int t = wave; t < 25; t += NWAVES) {
            int tm = t / 5, tn = t % 5;
            int row = tm * 16 + m;          // V row (M)
            int col = tn * 16 + m;          // U_a col (N)
            v8f acc = {0.f, 0.f, 0.f, 0.f, 0.f, 0.f, 0.f, 0.f};
            #pragma unroll 4
            for (int kc = 0; kc < 20; ++kc) {
                int k0 = kc * 4 + 2 * h;
                v2f af, bf;
                af.x = Vs[row * NP + k0];
                af.y = Vs[row * NP + k0 + 1];
                bf.x = trow[A2p[k0 * NP + col]];
                bf.y = trow[A2p[(k0 + 1) * NP + col]];
                acc = __builtin_amdgcn_wmma_f32_16x16x4_f32(
                          false, af, false, bf, (short)0, acc, false, false);
            }
            #pragma unroll
            for (int r = 0; r < 8; ++r)
                Xs[(tm * 16 + r + 8 * h) * NP + tn * 16 + m] = acc[r];
        }
        __syncthreads();

        // Phase B: Z_a = X_a * V^T ; contract Z_a[j,i] against (A1[i,j]==a)
        for (int t = wave; t < 25; t += NWAVES) {
            int tj = t / 5, ti = t % 5;
            int row  = tj * 16 + m;         // X row (j)
            int colv = ti * 16 + m;         // V row used as V^T col (i)
            v8f acc = {0.f, 0.f, 0.f, 0.f, 0.f, 0.f, 0.f, 0.f};
            #pragma unroll 4
            for (int kc = 0; kc < 20; ++kc) {
                int k0 = kc * 4 + 2 * h;
                v2f af, bf;
                af.x = Xs[row * NP + k0];
                af.y = Xs[row * NP + k0 + 1];
                bf.x = Vs[colv * NP + k0];
                bf.y = Vs[colv * NP + k0 + 1];
                acc = __builtin_amdgcn_wmma_f32_16x16x4_f32(
                          false, af, false, bf, (short)0, acc, false, false);
            }
            int i = ti * 16 + m;
            #pragma unroll
            for (int r = 0; r < 8; ++r) {
                int j = tj * 16 + r + 8 * h;
                int labv = (i < NG && j < NG) ? A1[i * NG + j] : 0;
                quad += (labv == a) ? acc[r] : 0.0f;
            }
        }
        __syncthreads();
    }

    // ---- diagonal correction (Dmat zeroes C's diagonal) and c . v ----
    float diag = 0.0f, cv = 0.0f;
    for (int idx = tid; idx < NN; idx += NTHREADS) {
        int i = idx / NP, j = idx % NP;     // i: graph-1 node, j: graph-2 node
        float v = Vs[idx];
        float cij;
        if (i < NG && j < NG)           cij = ncost[lab1[i] * NLAB + lab2[j]];
        else if (i == NG && j == NG)    cij = 0.0f;
        else                            cij = NODE_INS_DEL;
        cv += cij * v;
        int a1d = (i < NG) ? A1[i * NG + i] : 0;   // A1 diagonal label
        int a2d = (int)A2p[j * NP + j];
        diag += tableL[a1d * 5 + a2d] * v * v;
    }

    // ---- deterministic reduction: shfl within wave, then serial over waves ----
    for (int off = 16; off > 0; off >>= 1) {
        quad += __shfl_down(quad, off);
        diag += __shfl_down(diag, off);
        cv   += __shfl_down(cv,   off);
    }
    if (lane == 0) { wq[wave] = quad; wdg[wave] = diag; wcv[wave] = cv; }
    __syncthreads();
    if (tid == 0) {
        float Q = 0.f, D = 0.f, Cv = 0.f;
        for (int w = 0; w < NWAVES; ++w) { Q += wq[w]; D += wdg[w]; Cv += wcv[w]; }
        out[0] = 0.5f * (Q - D) + Cv;
    }
}

extern "C" void kernel_launch(void* const* d_in, const int* in_sizes, int n_in,
                              void* d_out, int out_size, void* d_ws, size_t ws_size,
                              hipStream_t stream) {
    (void)in_sizes; (void)n_in; (void)out_size; (void)d_ws; (void)ws_size;
    const int*   A1  = (const int*)d_in[0];
    const int*   A2  = (const int*)d_in[1];
    const int*   l1  = (const int*)d_in[2];
    const int*   l2  = (const int*)d_in[3];
    const float* nw  = (const float*)d_in[4];
    const float* ew  = (const float*)d_in[5];
    float*       out = (float*)d_out;
    ged_kernel<<<1, NTHREADS, 0, stream>>>(A1, A2, l1, l2, nw, ew, out);
}